// GAT_Attention_Block_91173565759952
// MI455X (gfx1250) — compile-verified
//
#include <hip/hip_runtime.h>

// GAT attention block, algebraically collapsed:
//   v1 = W_w^T u1, v2 = W_w^T u2, cc = W_b.(u1+u2)
//   mult[n,b] = dot(ai_sq[b], v1) + dot(ai_sn[n,b], v2) + cc
//   out[n,b]  = exp(leaky_relu(mult[n,b])) / sum_n exp(leaky_relu(mult[n,b]))
//
// Memory-bound (128 MiB of ai_sn dominates; ~6 us at 23.3 TB/s). Dot products
// use fp32 WMMA (V_WMMA_F32_16X16X4_F32): A = 16 batch rows x K=4 chunk,
// B = v-chunk replicated across the 16 columns, C accumulates over 32 chunks.
// B chunks are pinned in VGPRs via an empty asm so the hot loop issues exactly
// one global_load_b64 per WMMA (the unavoidable ai_sn stream).

#define IN_DIM  128
#define OUT_DIM 128
#define N_NEIGH 32
#define BATCH   8192
#define WAVES_PER_BLOCK 4

typedef float v2f __attribute__((ext_vector_type(2)));
typedef float v8f __attribute__((ext_vector_type(8)));

// ---------------------------------------------------------------------------
// Kernel 1: tiny precompute of v1[128], v2[128], cc  -> ws[0..256]
// ---------------------------------------------------------------------------
__global__ __launch_bounds__(IN_DIM) void gat_prep(const float* __restrict__ W_w,
                                                   const float* __restrict__ W_b,
                                                   const float* __restrict__ u,
                                                   float* __restrict__ ws) {
  const int i = threadIdx.x;  // 0..127, column of W_w
  float s1 = 0.0f, s2 = 0.0f;
  for (int o = 0; o < OUT_DIM; ++o) {
    const float w = W_w[o * IN_DIM + i];   // coalesced across i
    s1 = fmaf(u[o], w, s1);
    s2 = fmaf(u[OUT_DIM + o], w, s2);
  }
  ws[i] = s1;             // v1
  ws[IN_DIM + i] = s2;    // v2
  if (i == 0) {
    float c = 0.0f;
    for (int o = 0; o < OUT_DIM; ++o)
      c = fmaf(u[o] + u[OUT_DIM + o], W_b[o], c);
    ws[2 * IN_DIM] = c;   // cc
  }
}

// ---------------------------------------------------------------------------
// Kernel 2: main. One wave handles 16 batch rows. fp32 WMMA dot products.
// ---------------------------------------------------------------------------
__global__ __launch_bounds__(32 * WAVES_PER_BLOCK)
void gat_main(const float* __restrict__ ai_sq,
              const float* __restrict__ ai_sn,
              const float* __restrict__ ws,
              float* __restrict__ out) {
  __shared__ float denomLds[WAVES_PER_BLOCK][16];

  const int lane = threadIdx.x & 31;
  const int w    = threadIdx.x >> 5;
  const int tile = blockIdx.x * WAVES_PER_BLOCK + w;  // 0..511
  const int b0   = tile * 16;
  const int row  = lane & 15;                 // A-matrix M index held by this lane
  const int koff = (lane >> 4) * 2;           // K sub-offset: 0 (lanes 0-15) / 2 (lanes 16-31)

  const float* __restrict__ v1 = ws;
  const float* __restrict__ v2 = ws + IN_DIM;
  const float cc = ws[2 * IN_DIM];

  // --- Hoist B-matrix chunks (v2 replicated across 16 columns) -------------
  // B 4x16 f32 layout mirrors A: lane<16 holds B[0..1][n], lane>=16 B[2..3][n].
  // With replicated columns, lane's float2 is v[4c+koff .. 4c+koff+1].
  v2f bv2[32];
#pragma unroll
  for (int c = 0; c < 32; ++c)
    bv2[c] = *(const v2f*)(v2 + 4 * c + koff);
  // Pin in VGPRs: opaque defs prevent the compiler from re-loading these
  // inside the n-loop (it re-materialized them from global in round 1).
#pragma unroll
  for (int c = 0; c < 32; ++c)
    asm volatile("" : "+v"(bv2[c]));

  // --- acc = dot(ai_sq[b0+m], v1) + cc, for the 16 rows of this tile -------
  const float* sqRow = ai_sq + (size_t)(b0 + row) * IN_DIM + koff;
  v8f acc = {cc, cc, cc, cc, cc, cc, cc, cc};
#pragma unroll
  for (int c = 0; c < 32; ++c) {
    v2f a = *(const v2f*)(sqRow + 4 * c);
    v2f b = *(const v2f*)(v1 + 4 * c + koff);   // cold path, reloads are fine
    acc = __builtin_amdgcn_wmma_f32_16x16x4_f32(
        /*neg_a=*/false, a, /*neg_b=*/false, b,
        /*c_mod=*/(short)0, acc, /*reuse_a=*/false, /*reuse_b=*/false);
  }
  // D layout: lane 0 holds rows b0+0..7 in comps 0..7; lane 16 holds b0+8..15.

  v8f denom = {0.f, 0.f, 0.f, 0.f, 0.f, 0.f, 0.f, 0.f};

  for (int n = 0; n < N_NEIGH; ++n) {
    const float* snRow =
        ai_sn + ((size_t)n * BATCH + (size_t)(b0 + row)) * IN_DIM + koff;
    if (n + 1 < N_NEIGH)   // prefetch next neighbor row -> global_prefetch_b8
      __builtin_prefetch(
          ai_sn + ((size_t)(n + 1) * BATCH + (size_t)(b0 + row)) * IN_DIM + koff,
          0, 1);

    v8f c = acc;           // start from sq-dot + cc: C accumulates full mult
#pragma unroll
    for (int ck = 0; ck < 32; ++ck) {
      v2f a = *(const v2f*)(snRow + 4 * ck);   // the one real HBM stream
      c = __builtin_amdgcn_wmma_f32_16x16x4_f32(
          false, a, false, bv2[ck], (short)0, c, false, false);
    }

    // beta = exp(leaky_relu(mult)); accumulate denominator per component
    v8f e;
#pragma unroll
    for (int r = 0; r < 8; ++r) {
      float m = c[r];
      m = (m > 0.0f) ? m : 0.01f * m;
      float ev = __expf(m);
      e[r] = ev;
      denom[r] += ev;
    }

    // Lanes 0 and 16 own the 16 distinct rows -> two float4 stores each.
    if ((lane & 15) == 0) {
      float* o = out + (size_t)n * BATCH + b0 + (lane >> 4) * 8;
      float4 lo = make_float4(e[0], e[1], e[2], e[3]);
      float4 hi = make_float4(e[4], e[5], e[6], e[7]);
      *(float4*)(o + 0) = lo;
      *(float4*)(o + 4) = hi;
    }
  }

  // Transpose denominators through LDS: comp r of lane {0,16} -> row index.
  if ((lane & 15) == 0) {
#pragma unroll
    for (int r = 0; r < 8; ++r)
      denomLds[w][(lane >> 4) * 8 + r] = denom[r];
  }

  // Ensure beta stores have landed before the read-modify-write pass.
  asm volatile("s_wait_storecnt 0" ::: "memory");

  float rden = 0.0f;
  if (lane < 16) rden = 1.0f / denomLds[w][lane];

  // Normalization pass: coalesced 16-lane read/scale/write of out (1 MiB,
  // negligible vs the 128 MiB input stream).
  if (lane < 16) {
    for (int n = 0; n < N_NEIGH; ++n) {
      float* o = out + (size_t)n * BATCH + b0 + lane;
      *o = *o * rden;
    }
  }
}

// ---------------------------------------------------------------------------
extern "C" void kernel_launch(void* const* d_in, const int* in_sizes, int n_in,
                              void* d_out, int out_size, void* d_ws, size_t ws_size,
                              hipStream_t stream) {
  const float* ai_sq = (const float*)d_in[0];  // (8192,128)
  const float* ai_sn = (const float*)d_in[1];  // (32,8192,128)
  const float* W_w   = (const float*)d_in[2];  // (128,128)
  const float* W_b   = (const float*)d_in[3];  // (128,)
  const float* u     = (const float*)d_in[4];  // (256,)
  float* out = (float*)d_out;                  // (32,8192)
  float* ws  = (float*)d_ws;                   // v1[128] | v2[128] | cc

  gat_prep<<<dim3(1), dim3(IN_DIM), 0, stream>>>(W_w, W_b, u, ws);

  const int tiles  = BATCH / 16;                       // 512 waves
  const int blocks = tiles / WAVES_PER_BLOCK;          // 128 blocks x 128 thr
  gat_main<<<dim3(blocks), dim3(32 * WAVES_PER_BLOCK), 0, stream>>>(
      ai_sq, ai_sn, ws, out);
}